// RelaxedLogSumExp_3710851744391
// MI455X (gfx1250) — compile-verified
//
#include <hip/hip_runtime.h>
#include <math.h>

#define BDIM 128
#define NDIM 512
#define EDIM 512
#define KDIM 256
#define DD   1e-6f

typedef __attribute__((ext_vector_type(2))) float v2f;
typedef __attribute__((ext_vector_type(8))) float v8f;

// ---------------------------------------------------------------- kernel 1
// per-batch max, temp_head = head - max
__global__ void k1_max_sub(const float* __restrict__ head,
                           float* __restrict__ temp,
                           float* __restrict__ maxh) {
    int b = blockIdx.x, t = threadIdx.x;          // 256 threads
    __shared__ float red[256];
    float x0 = head[b * NDIM + t];
    float x1 = head[b * NDIM + t + 256];
    red[t] = fmaxf(x0, x1);
    __syncthreads();
    for (int s = 128; s > 0; s >>= 1) {
        if (t < s) red[t] = fmaxf(red[t], red[t + s]);
        __syncthreads();
    }
    float mx = red[0];
    temp[b * NDIM + t]       = x0 - mx;
    temp[b * NDIM + t + 256] = x1 - mx;
    if (t == 0) maxh[b] = mx;
}

// ---------------------------------------------------------------- kernel 2
// lower-median of |errs| over k=256 chunk, per (b, chunk, n).
// Tile (256 rows x 32 cols) is staged into LDS with async B128 copies
// (ASYNCcnt path, no VGPR round-trip), then bitonic-sorted keyed on |x|
// (abs folds into the compare as a free VALU source modifier).
__global__ void k2_median(const float* __restrict__ errs,
                          float* __restrict__ med) {
    __shared__ float s[256 * 32];                 // 32 KB
    int blk  = blockIdx.x;                        // B*2*16 = 4096 blocks
    int nt   = blk & 15;
    int c    = (blk >> 4) & 1;
    int b    = blk >> 5;
    int t    = threadIdx.x;
    int lane = t & 31, w = t >> 5;
    int n0   = nt * 32;
    const float* base = errs + ((size_t)(b * EDIM + c * KDIM)) * NDIM + n0;

    // async global->LDS: lane l covers row (l>>3), 16 contiguous bytes at
    // col 4*(l&7); one wave-op moves 4 rows (512 B), 8 ops cover 32 rows/wave.
    unsigned ldsbase = (unsigned)(uintptr_t)(&s[0]);
    int r0 = lane >> 3, colq = lane & 7;
#pragma unroll
    for (int i = 0; i < 8; ++i) {
        int j = w * 32 + i * 4 + r0;
        unsigned long long gaddr =
            (unsigned long long)(uintptr_t)(base + (size_t)j * NDIM + colq * 4);
        unsigned laddr = ldsbase + (unsigned)((j * 32 + colq * 4) * 4);
        asm volatile("global_load_async_to_lds_b128 %0, %1, off"
                     :: "v"(laddr), "v"(gaddr) : "memory");
    }
    asm volatile("s_wait_asynccnt 0x0" ::: "memory");
    __syncthreads();

    // bitonic sort of 256 per column (raw values, |x| key); lane = column so
    // LDS is bank-conflict-free and the i-branch is wave-uniform.
    for (int k = 2; k <= 256; k <<= 1) {
        for (int j = k >> 1; j > 0; j >>= 1) {
            for (int ii = 0; ii < 32; ++ii) {
                int i = w * 32 + ii;
                int ixj = i ^ j;
                if (ixj > i) {
                    float x = s[i * 32 + lane];
                    float y = s[ixj * 32 + lane];
                    bool up = ((i & k) == 0);
                    if ((fabsf(x) > fabsf(y)) == up) {
                        s[i * 32 + lane] = y;
                        s[ixj * 32 + lane] = x;
                    }
                }
            }
            __syncthreads();
        }
    }
    if (t < 32)
        med[((size_t)(b * 2 + c)) * NDIM + n0 + t] =
            fabsf(s[127 * 32 + t]);               // rank (256-1)/2
}

// ---------------------------------------------------------------- kernel 3
// exp relaxation: lam/delta per (b,n), exp_head[b] = sum_n(h*lam+mu)
__global__ void k3_relax_exp(const float* __restrict__ temp,
                             const float* __restrict__ med,
                             float* __restrict__ lam,
                             float* __restrict__ delta,
                             float* __restrict__ exph) {
    int b = blockIdx.x, t = threadIdx.x;          // 256 threads
    __shared__ float red[256];
    float acc = 0.f;
    for (int u = 0; u < 2; ++u) {
        int n    = t + u * 256;
        float h  = temp[b * NDIM + n];
        float l1 = med[(size_t)(b * 2 + 0) * NDIM + n] +
                   med[(size_t)(b * 2 + 1) * NDIM + n];
        float lb = h - l1, ub = h + l1;
        bool tight = (ub == lb);
        float elb = expf(lb), eub = expf(ub);
        float lam_s = (eub - elb) / (ub - lb + DD);
        float lm  = fminf(lam_s, expf(lb + 0.9f));
        float ub0 = (lm > lam_s) ? (elb - lm * lb) : (eub - lm * ub);
        float t1  = lm * (1.0f - logf(lm));
        float mu  = 0.5f * (t1 + ub0);
        float dl  = 0.5f * (ub0 - t1);
        if (tight) { lm = 0.f; mu = eub; dl = 0.f; }
        lam[b * NDIM + n]   = lm;
        delta[b * NDIM + n] = dl;
        acc += h * lm + mu;
    }
    red[t] = acc;
    __syncthreads();
    for (int s = 128; s > 0; s >>= 1) {
        if (t < s) red[t] += red[t + s];
        __syncthreads();
    }
    if (t == 0) exph[b] = red[0];
}

// ---------------------------------------------------------------- kernel 4
// exp_errs[b,e] = sum_n src[b,e,n] * vec[b,n]  via V_WMMA_F32_16X16X4_F32.
// One wave per 16-row tile. A = src tile (16x4), B = vec segment in col 0.
// Software-pipelined: 16 loads (8 WMMA-steps worth) issued into distinct
// registers before the dependent WMMA chain, so each wave keeps many
// loads in flight instead of one s_wait_loadcnt 0 round-trip per step.
__global__ void k4_matvec_wmma(const float* __restrict__ errs,
                               const float* __restrict__ cexp,
                               const float* __restrict__ lam,
                               const float* __restrict__ delta,
                               float* __restrict__ eerr) {
    int blk  = blockIdx.x;                        // B * 12 = 1536 blocks
    int b    = blk / 12, g = blk % 12;
    int t    = threadIdx.x;                       // 128 threads = 4 waves
    int w    = t >> 5, lane = t & 31;
    int e0   = (g * 4 + w) * 16;
    int m    = lane & 15;
    int half = lane >> 4;

    const float* rowp;
    const float* vecp;
    if (e0 < EDIM) {                              // errs part (rows 0..511)
        rowp = errs + ((size_t)b * EDIM + e0 + m) * NDIM;
        vecp = lam + (size_t)b * NDIM;
    } else {                                      // cauchy_exp part (rows 512..767)
        rowp = cexp + ((size_t)b * KDIM + (e0 - EDIM) + m) * NDIM;
        vecp = delta + (size_t)b * NDIM;
    }
    rowp += 2 * half;
    vecp += 2 * half;

    v8f c = {};
#pragma unroll
    for (int g8 = 0; g8 < NDIM / 32; ++g8) {      // 16 groups of 8 WMMA steps
        int nbase = g8 * 32;
        if ((g8 & 1) == 0)
            __builtin_prefetch(rowp + nbase + 64, 0, 1);  // global_prefetch_b8

        v2f areg[8];
        v2f breg[8];
#pragma unroll
        for (int q = 0; q < 8; ++q) {             // 16 loads, no waits between
            areg[q] = *(const v2f*)(rowp + nbase + q * 4);
            breg[q] = *(const v2f*)(vecp + nbase + q * 4);
        }
#pragma unroll
        for (int q = 0; q < 8; ++q) {
            // B 4x16 col 0 only: lane 0 -> K0,K1 ; lane 16 -> K2,K3 ; rest 0.
            v2f bv;
            bv.x = (m == 0) ? breg[q].x : 0.f;
            bv.y = (m == 0) ? breg[q].y : 0.f;
            c = __builtin_amdgcn_wmma_f32_16x16x4_f32(false, areg[q], false, bv,
                                                      (short)0, c, false, false);
        }
    }

    // D column 0: lane 0 holds M=0..7 (vgpr 0..7), lane 16 holds M=8..15
    float* outp = eerr + (size_t)b * 768 + e0;
    if (lane == 0)  { for (int r = 0; r < 8; ++r) outp[r]     = c[r]; }
    if (lane == 16) { for (int r = 0; r < 8; ++r) outp[8 + r] = c[r]; }
}

// ---------------------------------------------------------------- kernel 5
// medians of exp_errs (3 chunks of 256), log relaxation, final outputs.
__global__ void k5_relax_log(const float* __restrict__ eerr,
                             const float* __restrict__ exph,
                             const float* __restrict__ maxh,
                             const float* __restrict__ clog,
                             float* __restrict__ out) {
    int b = blockIdx.x, t = threadIdx.x;          // 256 threads
    __shared__ float a[3 * 256];
    __shared__ float orig[3 * 256];
    for (int c3 = 0; c3 < 3; ++c3) {
        float v = eerr[(size_t)b * 768 + c3 * 256 + t];
        orig[c3 * 256 + t] = v;
        a[c3 * 256 + t]    = fabsf(v);
    }
    __syncthreads();
    for (int c3 = 0; c3 < 3; ++c3) {
        float* s = a + c3 * 256;
        for (int k = 2; k <= 256; k <<= 1)
            for (int j = k >> 1; j > 0; j >>= 1) {
                int i = t, ixj = t ^ j;
                if (ixj > i) {
                    float x = s[i], y = s[ixj];
                    bool up = ((i & k) == 0);
                    if ((x > y) == up) { s[i] = y; s[ixj] = x; }
                }
                __syncthreads();
            }
    }
    float l1 = a[127] + a[256 + 127] + a[512 + 127];
    float eh = exph[b];
    float lb = eh - l1, ub = eh + l1;
    bool tight = (ub == lb);
    float lam_s = (logf(ub) - logf(lb + DD)) / (ub - lb + DD);
    float lm  = lam_s;
    float lb0 = (lm < lam_s) ? (logf(ub) - ub * lm) : (logf(lb + DD) - lb * lm);
    float nll = -logf(lm) - 1.0f;
    float mu  = 0.5f * (nll + lb0);
    float dl  = 0.5f * (nll - lb0);
    if (tight) { lm = 0.f; mu = logf(ub); dl = 0.f; }

    float* lse = out + BDIM + (size_t)b * 1024;
    for (int c3 = 0; c3 < 3; ++c3)
        lse[c3 * 256 + t] = orig[c3 * 256 + t] * lm;
    lse[768 + t] = dl * clog[b * 256 + t];
    if (t == 0) out[b] = eh * lm + mu + maxh[b];
}

// ---------------------------------------------------------------- launcher
extern "C" void kernel_launch(void* const* d_in, const int* in_sizes, int n_in,
                              void* d_out, int out_size, void* d_ws, size_t ws_size,
                              hipStream_t stream) {
    const float* head = (const float*)d_in[0];    // (B,N)
    const float* errs = (const float*)d_in[1];    // (B*E,N)
    const float* cexp = (const float*)d_in[2];    // (B,K,N)
    const float* clog = (const float*)d_in[3];    // (B,K)
    float* ws = (float*)d_ws;
    float* temp  = ws;            // 65536
    float* maxh  = ws + 65536;    // 128
    float* med   = ws + 65664;    // 131072  (B,2,N)
    float* lam   = ws + 196736;   // 65536
    float* delta = ws + 262272;   // 65536
    float* exph  = ws + 327808;   // 128
    float* eerr  = ws + 327936;   // 98304   (B,768)
    float* out   = (float*)d_out; // 128 + 131072

    k1_max_sub    <<<BDIM, 256, 0, stream>>>(head, temp, maxh);
    k2_median     <<<BDIM * 2 * (NDIM / 32), 256, 0, stream>>>(errs, med);
    k3_relax_exp  <<<BDIM, 256, 0, stream>>>(temp, med, lam, delta, exph);
    k4_matvec_wmma<<<BDIM * 12, 128, 0, stream>>>(errs, cexp, lam, delta, eerr);
    k5_relax_log  <<<BDIM, 256, 0, stream>>>(eerr, exph, maxh, clog, out);
}